// E3nnModel_58583353917863
// MI455X (gfx1250) — compile-verified
//
#include <hip/hip_runtime.h>

typedef __attribute__((ext_vector_type(2))) float v2f;
typedef __attribute__((ext_vector_type(8))) float v8f;

#define C01 0.57735026918962576451f   /* 1/sqrt(3) */
#define C11 0.40824829046386301637f   /* 1/sqrt(6) */
#define PW2 1.22474487139158904910f   /* sqrt(1.5) */

__global__ void zero_out_kernel(float* __restrict__ out, int n) {
    int i = blockIdx.x * blockDim.x + threadIdx.x;
    if (i < n) out[i] = 0.0f;
}

__device__ __forceinline__ void tp_step(float w0, float w1, float s,
                                        float vx, float vy, float vz,
                                        float& ox, float& oy, float& oz) {
    float a = PW2 * w0 * C01 * s;
    float b = PW2 * w1 * C11;
    float cx = oy * vz - oz * vy;
    float cy = oz * vx - ox * vz;
    float cz = ox * vy - oy * vx;
    float nx = a * ox + b * cx;
    float ny = a * oy + b * cy;
    float nz = a * oz + b * cz;
    ox = nx; oy = ny; oz = nz;
}

// Per-lane chain up to the pre-fc "out" vector.
__device__ __forceinline__ void node_chain(
    int i,
    const int* __restrict__ z, const float* __restrict__ pos,
    const float* __restrict__ embed_table,
    const float* __restrict__ lin_w, const float* __restrict__ lin_b,
    const float* __restrict__ tp1_w, const float* __restrict__ tp2_w,
    const float* __restrict__ tp3_w, const float* __restrict__ gate_w,
    float& ox, float& oy, float& oz) {
    float s  = embed_table[z[i]];
    float px = pos[3 * i + 0], py = pos[3 * i + 1], pz = pos[3 * i + 2];

    float vx = lin_w[0] * px + lin_w[1] * py + lin_w[2] * pz + lin_b[0];
    float vy = lin_w[3] * px + lin_w[4] * py + lin_w[5] * pz + lin_b[1];
    float vz = lin_w[6] * px + lin_w[7] * py + lin_w[8] * pz + lin_b[2];

    // layer 1: cross(v,v)==0 structurally, PW1==1
    float k1 = (tp1_w[0] + tp1_w[1]) * C01 * s;
    ox = k1 * vx; oy = k1 * vy; oz = k1 * vz;

    tp_step(tp2_w[0], tp2_w[1], s, vx, vy, vz, ox, oy, oz);
    tp_step(tp3_w[0], tp3_w[1], s, vx, vy, vz, ox, oy, oz);
    tp_step(gate_w[0], gate_w[1], s, vx, vy, vz, ox, oy, oz);
}

// Main fused kernel: 256 nodes / block (8 wave32), N assumed multiple of 256.
__global__ __launch_bounds__(256) void e3nn_fused_kernel(
    const int* __restrict__ z, const float* __restrict__ pos,
    const int* __restrict__ batch, const float* __restrict__ embed_table,
    const float* __restrict__ lin_w, const float* __restrict__ lin_b,
    const float* __restrict__ tp1_w, const float* __restrict__ tp2_w,
    const float* __restrict__ tp3_w, const float* __restrict__ gate_w,
    const float* __restrict__ fc_w, const float* __restrict__ fc_b,
    float* __restrict__ out) {
    const int tid = threadIdx.x;
    const int i   = blockIdx.x * 256 + tid;

    float ox, oy, oz;
    node_chain(i, z, pos, embed_table, lin_w, lin_b, tp1_w, tp2_w, tp3_w,
               gate_w, ox, oy, oz);
    const int bi = batch[i];

    // ---- fc via V_WMMA_F32_16X16X4_F32 ----
    // D(16x16) = A(16x4) x B(4x16):
    //   A = fc_w padded: row M = output component, col K = input component.
    //   B column n = pre-fc "out" of node n (16 nodes per WMMA).
    const int  lane  = tid & 31;
    const int  l15   = lane & 15;
    const bool upper = lane >= 16;

    // A layout (f32 16x4): lanes 0-15 -> {v0:K0, v1:K1}; lanes 16-31 -> {v0:K2, v1:K3}
    v2f A;
    A[0] = (l15 < 3) ? (upper ? fc_w[l15 * 3 + 2] : fc_w[l15 * 3 + 0]) : 0.0f;
    A[1] = (l15 < 3 && !upper) ? fc_w[l15 * 3 + 1] : 0.0f;

    // Cross-lane staging for B (B rows stripe across lanes: v0={K0 lo, K2 hi}, v1={K1 lo, K3 hi})
    float oz_lo = __shfl(oz, l15, 32);       // oz of node (l15)        -> upper half of B1.v0
    float ox_hi = __shfl(ox, l15 + 16, 32);  // ox of node (16+l15)     -> lower half of B2.v0
    float oy_hi = __shfl(oy, l15 + 16, 32);  // oy of node (16+l15)     -> lower half of B2.v1

    v2f B1, B2;
    B1[0] = upper ? oz_lo : ox;   // WMMA#1: nodes 0..15 of this wave
    B1[1] = upper ? 0.0f  : oy;
    B2[0] = upper ? oz    : ox_hi; // WMMA#2: nodes 16..31 of this wave
    B2[1] = upper ? 0.0f  : oy_hi;

    v8f c0 = {0.f, 0.f, 0.f, 0.f, 0.f, 0.f, 0.f, 0.f};
    v8f D1 = __builtin_amdgcn_wmma_f32_16x16x4_f32(false, A, false, B1,
                                                   (short)0, c0, false, false);
    v8f D2 = __builtin_amdgcn_wmma_f32_16x16x4_f32(false, A, false, B2,
                                                   (short)0, c0, false, false);

    // D rows 0..2 (= y components) live in v0..v2, columns across lanes 0-15.
    float d2x = __shfl(D2[0], l15, 32);
    float d2y = __shfl(D2[1], l15, 32);
    float d2z = __shfl(D2[2], l15, 32);
    float yx = (upper ? d2x : D1[0]) + fc_b[0];
    float yy = (upper ? d2y : D1[1]) + fc_b[1];
    float yz = (upper ? d2z : D1[2]) + fc_b[2];

    // ---- segmented reduction (batch is sorted) ----
    __shared__ float sx[256], sy[256], sz[256];
    __shared__ int   sseg[256];
    sx[tid] = yx; sy[tid] = yy; sz[tid] = yz; sseg[tid] = bi;
    __syncthreads();

    bool head = (tid == 0) || (sseg[tid - 1] != bi);
    if (head) {
        float ax = 0.f, ay = 0.f, az = 0.f;
        int j = tid;
        while (j < 256 && sseg[j] == bi) {
            ax += sx[j]; ay += sy[j]; az += sz[j]; ++j;
        }
        atomicAdd(&out[3 * bi + 0], ax);
        atomicAdd(&out[3 * bi + 1], ay);
        atomicAdd(&out[3 * bi + 2], az);
    }
}

// Scalar tail for N % 256 != 0 (not hit for N = 4,194,304; keeps EXEC-all-ones
// guarantee intact in the WMMA kernel).
__global__ void e3nn_tail_kernel(
    const int* __restrict__ z, const float* __restrict__ pos,
    const int* __restrict__ batch, const float* __restrict__ embed_table,
    const float* __restrict__ lin_w, const float* __restrict__ lin_b,
    const float* __restrict__ tp1_w, const float* __restrict__ tp2_w,
    const float* __restrict__ tp3_w, const float* __restrict__ gate_w,
    const float* __restrict__ fc_w, const float* __restrict__ fc_b,
    float* __restrict__ out, int start, int n) {
    int i = start + blockIdx.x * blockDim.x + threadIdx.x;
    if (i >= n) return;
    float ox, oy, oz;
    node_chain(i, z, pos, embed_table, lin_w, lin_b, tp1_w, tp2_w, tp3_w,
               gate_w, ox, oy, oz);
    float yx = fc_w[0] * ox + fc_w[1] * oy + fc_w[2] * oz + fc_b[0];
    float yy = fc_w[3] * ox + fc_w[4] * oy + fc_w[5] * oz + fc_b[1];
    float yz = fc_w[6] * ox + fc_w[7] * oy + fc_w[8] * oz + fc_b[2];
    int bi = batch[i];
    atomicAdd(&out[3 * bi + 0], yx);
    atomicAdd(&out[3 * bi + 1], yy);
    atomicAdd(&out[3 * bi + 2], yz);
}

extern "C" void kernel_launch(void* const* d_in, const int* in_sizes, int n_in,
                              void* d_out, int out_size, void* d_ws, size_t ws_size,
                              hipStream_t stream) {
    const int*   z      = (const int*)d_in[0];
    const float* pos    = (const float*)d_in[1];
    const int*   batch  = (const int*)d_in[2];
    const float* embed  = (const float*)d_in[3];
    const float* lin_w  = (const float*)d_in[4];
    const float* lin_b  = (const float*)d_in[5];
    const float* tp1_w  = (const float*)d_in[6];
    const float* tp2_w  = (const float*)d_in[7];
    const float* tp3_w  = (const float*)d_in[8];
    const float* gate_w = (const float*)d_in[9];
    const float* fc_w   = (const float*)d_in[10];
    const float* fc_b   = (const float*)d_in[11];
    float* out = (float*)d_out;

    const int N = in_sizes[0];

    // d_out is poisoned by the harness: zero it first (same stream => ordered).
    zero_out_kernel<<<(out_size + 255) / 256, 256, 0, stream>>>(out, out_size);

    const int nfull = N / 256;            // full 256-node blocks (WMMA path)
    if (nfull > 0) {
        e3nn_fused_kernel<<<nfull, 256, 0, stream>>>(
            z, pos, batch, embed, lin_w, lin_b, tp1_w, tp2_w, tp3_w, gate_w,
            fc_w, fc_b, out);
    }
    const int rem = N - nfull * 256;
    if (rem > 0) {
        e3nn_tail_kernel<<<(rem + 255) / 256, 256, 0, stream>>>(
            z, pos, batch, embed, lin_w, lin_b, tp1_w, tp2_w, tp3_w, gate_w,
            fc_w, fc_b, out, nfull * 256, N);
    }
}